// PolynomialADMRSolver_88304527606131
// MI455X (gfx1250) — compile-verified
//
#include <hip/hip_runtime.h>

// CDNA5 / gfx1250 — wave32, WMMA (not MFMA).
typedef float v2f __attribute__((ext_vector_type(2)));
typedef float v8f __attribute__((ext_vector_type(8)));

#define B_DIM 4096
#define N_DIM 32
#define D_DIM 1024

// One block per batch. 256 threads = 8 wave32 waves.
// Each wave computes 8 d-tiles of width 16 using V_WMMA_F32_16X16X4_F32:
//   A (16x4)  : all 16 rows = adjacency-weight chunk w[k..k+3]  (broadcast rows)
//   B (4x16)  : neighbor_states[b, k..k+3, d0..d0+15]
//   C += A*B  : every row of the 16x16 tile equals the weighted neighbor sum
// A-layout (32-bit A 16x4): lane L holds M=L%16; VGPR0 -> K = (L<16?0:2),
// VGPR1 -> K = (L<16?1:3). B-layout: VGPR0 lanes0-15 = row K (cols 0-15),
// lanes16-31 = row K+2; VGPR1 = rows K+1, K+3. D-layout: VGPR0 lane L holds
// (M = L<16?0:8, N = L%16) -> all lanes carry out[d0 + L%16].
__global__ __launch_bounds__(256)
void poly_admr_kernel(const float* __restrict__ states,    // [B, D]
                      const float* __restrict__ nbr,       // [B, N, D]
                      const float* __restrict__ adj,       // [B, N]
                      const float* __restrict__ valence,   // [B]
                      float* __restrict__ out)             // [B, D]
{
    const int b    = blockIdx.x;
    const int lane = threadIdx.x & 31;
    const int wave = threadIdx.x >> 5;     // 0..7
    const int half = lane >> 4;            // 0 or 1 (lane half)
    const int m    = lane & 15;            // column within 16-wide d tile

    __shared__ float s_w[N_DIM];
    if (threadIdx.x < N_DIM)
        s_w[threadIdx.x] = adj[(size_t)b * N_DIM + threadIdx.x];
    __syncthreads();

    const float  val = valence[b];
    const float* Xb  = nbr    + (size_t)b * N_DIM * D_DIM;  // [32][1024]
    const float* Sb  = states + (size_t)b * D_DIM;
    float*       Ob  = out    + (size_t)b * D_DIM;

    #pragma unroll 2
    for (int i = 0; i < 8; ++i) {
        const int d0 = (i * 8 + wave) * 16;                 // 16-wide d tile

        // Prefetch next tile's first rows (emits global_prefetch_b8).
        if (i < 7) {
            const int dn = ((i + 1) * 8 + wave) * 16;
            __builtin_prefetch(Xb + (size_t)(2 * half) * D_DIM + dn + m, 0, 0);
            __builtin_prefetch(Xb + (size_t)(16 + 2 * half) * D_DIM + dn + m, 0, 0);
        }

        v8f c = {0.f, 0.f, 0.f, 0.f, 0.f, 0.f, 0.f, 0.f};

        #pragma unroll
        for (int k = 0; k < N_DIM; k += 4) {
            const int k0 = k + 2 * half;                    // rows this half owns
            // A: broadcast weight chunk into all 16 rows.
            v2f a;
            a.x = s_w[k0];
            a.y = s_w[k0 + 1];
            // B: neighbor tile rows k0, k0+1 at columns d0..d0+15.
            v2f bb;
            bb.x = Xb[(size_t)k0 * D_DIM + d0 + m];
            bb.y = Xb[(size_t)(k0 + 1) * D_DIM + d0 + m];
            // 8 args: (neg_a, A, neg_b, B, c_mod, C, reuse_a, reuse_b)
            c = __builtin_amdgcn_wmma_f32_16x16x4_f32(
                    false, a, false, bb, (short)0, c, false, false);
        }

        // All D rows are identical; VGPR0 lane L = weighted sum for d0 + (L%16).
        const float ws = c[0];

        // Gate + co-prime polynomial mean: (x^2 + x^3 + x^5 + x^7) / 4.
        const float x  = Sb[d0 + m] * (ws + val);
        const float x2 = x * x;
        const float x3 = x2 * x;
        const float x5 = x3 * x2;
        const float x7 = x5 * x2;
        const float r  = 0.25f * (x2 + x3 + x5 + x7);

        if (half == 0)                                      // lanes 16-31 duplicate
            Ob[d0 + m] = r;
    }
}

extern "C" void kernel_launch(void* const* d_in, const int* in_sizes, int n_in,
                              void* d_out, int out_size, void* d_ws, size_t ws_size,
                              hipStream_t stream) {
    const float* states  = (const float*)d_in[0];   // [B, D]
    const float* nbr     = (const float*)d_in[1];   // [B, N, D]
    const float* adj     = (const float*)d_in[2];   // [B, N]
    const float* valence = (const float*)d_in[3];   // [B]
    float*       out     = (float*)d_out;           // [B, D]

    const int B = in_sizes[3];                      // 4096
    poly_admr_kernel<<<dim3(B), dim3(256), 0, stream>>>(states, nbr, adj, valence, out);
}